// HMlr_70411693850776
// MI455X (gfx1250) — compile-verified
//
#include <hip/hip_runtime.h>
#include <cmath>

// Poincare MLR (c = 1):  out = 2*zn * asinh( lam*(x@z)*cosh(2r)/zn - (lam-1)*sinh(2r) )
// lam = 2 / (1 - ||x||^2),  zn = max(||z_col||, 1e-15)
//
// GEMM via V_WMMA_F32_16X16X4_F32 (fp32-exact matrix path on gfx1250).
// z is staged in LDS pre-swizzled so each B fragment {z[k][col], z[k+1][col]}
// is a single 8B word -> one ds_load_b64 straight into the WMMA operand pair,
// with K-pair row stride 288 dwords (== 32 mod 64 banks) so the two half-waves
// hit disjoint bank halves.

typedef __attribute__((ext_vector_type(2))) float v2f;
typedef __attribute__((ext_vector_type(8))) float v8f;

#if defined(__gfx1250__)
#if !__has_builtin(__builtin_amdgcn_wmma_f32_16x16x4_f32)
#error "gfx1250: __builtin_amdgcn_wmma_f32_16x16x4_f32 not available"
#endif
#endif

constexpr int kD = 128;                 // d_in == d_out == 128
constexpr int kRowsPerWave = 16;
constexpr int kWavesPerBlock = 8;
constexpr int kRowsPerBlock = kRowsPerWave * kWavesPerBlock;   // 128
constexpr int kPairStride = 288;        // dwords per K-pair row (288 % 64 == 32)

__global__ __launch_bounds__(256) void hmlr_poincare_mlr_wmma(
    const float* __restrict__ x,   // [N, 128]
    const float* __restrict__ z,   // [128, 128] row-major: z[k][col]
    const float* __restrict__ r,   // [128]
    float* __restrict__ out)       // [N, 128]
{
  // zSwz[p*288 + col*2 + (k&1)] = z[k][col], p = k>>1   (~72 KB)
  __shared__ float zSwz[(kD / 2) * kPairStride];
  __shared__ float znLds[kD];      // ||z_col||          (output scale 2*zn)
  __shared__ float caLds[kD];      // cosh(2r)/zn        (fused inner scale)
  __shared__ float shLds[kD];      // sinh(2r)

  const int tid  = threadIdx.x;
  const int wave = tid >> 5;
  const int lane = tid & 31;

  // ---- Phase 1: stage z into LDS, swizzled to K-pair-interleaved layout ----
  {
    #pragma unroll
    for (int i = 0; i < (kD * kD / 4) / 256; ++i) {
      const int idx = tid + i * 256;              // float4 index
      const float4 v = reinterpret_cast<const float4*>(z)[idx];
      const int k = idx >> 5;                     // row (128 floats = 32 float4)
      const int c = (idx & 31) << 2;              // starting col
      const int base = (k >> 1) * kPairStride + (k & 1);
      zSwz[base + (c + 0) * 2] = v.x;
      zSwz[base + (c + 1) * 2] = v.y;
      zSwz[base + (c + 2) * 2] = v.z;
      zSwz[base + (c + 3) * 2] = v.w;
    }
  }
  __syncthreads();

  // ---- Phase 2: per-output-column constants ----
  if (tid < kD) {
    float s = 0.f;
    #pragma unroll 8
    for (int p = 0; p < kD / 2; ++p) {
      const v2f v = *reinterpret_cast<const v2f*>(&zSwz[p * kPairStride + tid * 2]);
      s = fmaf(v.x, v.x, s);
      s = fmaf(v.y, v.y, s);
    }
    const float zn = fmaxf(sqrtf(s), 1e-15f);
    const float t2 = 2.0f * r[tid];               // cs = sqrt(C) = 1
    znLds[tid] = zn;
    caLds[tid] = coshf(t2) / zn;
    shLds[tid] = sinhf(t2);
  }
  __syncthreads();

  // ---- Phase 3: WMMA GEMM, 16 rows x 128 cols per wave, K = 128 ----
  const int rowBase = blockIdx.x * kRowsPerBlock + wave * kRowsPerWave;
  const int mOwn  = lane & 15;          // A/B layout: lane -> M / N
  const int half  = lane >> 4;          // 0: K{0,1}, 1: K{2,3}
  const int kHalf = half << 1;
  const float* xRow = x + (size_t)(rowBase + mOwn) * kD;
  // LDS base for this lane's B fragments: p = 2*ks + half, col = n*16 + mOwn
  const float* bBase = &zSwz[half * kPairStride + mOwn * 2];

  v8f acc[8] = {};                      // 8 N-tiles of 16x16 f32
  float rowsq = 0.f;                    // partial ||x_row||^2

  v2f a = *reinterpret_cast<const v2f*>(xRow + kHalf);     // pipelined A
  #pragma unroll
  for (int ks = 0; ks < kD / 4; ++ks) {
    const v2f aCur = a;
    if (ks + 1 < kD / 4)                // prefetch next A fragment
      a = *reinterpret_cast<const v2f*>(xRow + ((ks + 1) << 2) + kHalf);
    rowsq = fmaf(aCur.x, aCur.x, rowsq);
    rowsq = fmaf(aCur.y, aCur.y, rowsq);
    const float* bRow = bBase + (2 * ks) * kPairStride;    // p = 2*ks + half
    #pragma unroll
    for (int n = 0; n < 8; ++n) {
      const v2f b = *reinterpret_cast<const v2f*>(bRow + n * 32);  // one ds_load_b64
      acc[n] = __builtin_amdgcn_wmma_f32_16x16x4_f32(
          /*neg_a=*/false, aCur, /*neg_b=*/false, b,
          /*c_mod=*/(short)0, acc[n], /*reuse_a=*/false, /*reuse_b=*/false);
    }
  }

  // ---- lambda per row: combine the two K-halves (wave32 xor-16) ----
  rowsq += __shfl_xor(rowsq, 16, 32);
  const float lamOwn = 2.0f / (1.0f - rowsq);   // lane L holds lam(row L%16)

  // ---- Epilogue: C/D layout: VGPR g -> M = g + 8*(lane>=16); N = lane&15 ----
  const int mHalfRows = half << 3;              // 0 or 8
  #pragma unroll
  for (int g = 0; g < 8; ++g) {
    const float lam = __shfl(lamOwn, g + mHalfRows, 32);
    const float lm1 = lam - 1.0f;
    float* orow = out + (size_t)(rowBase + g + mHalfRows) * kD;
    #pragma unroll
    for (int n = 0; n < 8; ++n) {
      const int col = (n << 4) + mOwn;
      const float arg = fmaf(lam * acc[n][g], caLds[col], -lm1 * shLds[col]);
      orow[col] = 2.0f * znLds[col] * asinhf(arg);
    }
  }
}

extern "C" void kernel_launch(void* const* d_in, const int* in_sizes, int n_in,
                              void* d_out, int out_size, void* d_ws, size_t ws_size,
                              hipStream_t stream) {
  const float* x = (const float*)d_in[0];
  const float* z = (const float*)d_in[1];
  const float* r = (const float*)d_in[2];
  float* out = (float*)d_out;

  const int rows = in_sizes[0] / kD;            // 16*8192 = 131072
  const int grid = rows / kRowsPerBlock;        // 1024
  hipLaunchKernelGGL(hmlr_poincare_mlr_wmma, dim3(grid), dim3(256), 0, stream,
                     x, z, r, out);
}